// KVCache_9526237462719
// MI455X (gfx1250) — compile-verified
//
#include <hip/hip_runtime.h>
#include <hip/hip_bf16.h>

// Problem constants (from reference): B=4, H=32, BLOCK=4096, D=128, S=1024, fp32.
#define NB      4
#define NH      32
#define NBLOCK  4096
#define ND      128
#define NS      1024

#define BH         (NB * NH)                       // 128 batch*head rows
#define ROWS       ((size_t)BH * NS)               // 131072 output rows per tensor
#define ELEMS      (ROWS * ND)                     // 16,777,216 floats per tensor
#define Q_PER_ROW  (ND / 4)                        // 32 float4 per 512B row == one wave32

typedef __attribute__((ext_vector_type(4))) float f32x4;

// ---------------------------------------------------------------------------
// Pass 0a: inv[s] = -1 for s in [0, S)
// ---------------------------------------------------------------------------
__global__ void kv_init_inv(int* __restrict__ inv) {
    int i = blockIdx.x * blockDim.x + threadIdx.x;
    if (i < NS) inv[i] = -1;
}

// ---------------------------------------------------------------------------
// Pass 0b: inverse scatter map. inv[input_pos[s]] = s (last-s wins via
// atomicMax; positions >= S fall outside the returned slice -> dropped).
// ---------------------------------------------------------------------------
__global__ void kv_build_inv(const int* __restrict__ input_pos, int* __restrict__ inv) {
    int s = blockIdx.x * blockDim.x + threadIdx.x;
    if (s < NS) {
        int p = input_pos[s];
        if ((unsigned)p < (unsigned)NS) atomicMax(&inv[p], s);
    }
}

// ---------------------------------------------------------------------------
// Pass 1: fused one-touch gather, K and V per wave.
//
// One wave32 owns one (bh, s) row slot; lane l moves float4 #l of the 512B
// row for BOTH tensors: 2x global_load_b128 + 2x global_store_b128, all with
// non-temporal (TH_NT) policy — the 256 MiB stream is single-touch and larger
// than the 192 MB L2, so bypassing retention is the right gfx1250 cache hint.
// inv[s] is wave-uniform -> scalar load; the val/cache select is uniform too,
// so no divergence anywhere.
// ---------------------------------------------------------------------------
__global__ __launch_bounds__(256) void kv_gather_rows(
    const float* __restrict__ k_val,   const float* __restrict__ v_val,
    const float* __restrict__ k_cache, const float* __restrict__ v_cache,
    const int*   __restrict__ inv,     float* __restrict__ out)
{
    const unsigned t     = blockIdx.x * 256u + threadIdx.x;  // [0, ROWS*32)
    const unsigned row   = t >> 5;                           // output row (wave-uniform)
    const unsigned lane4 = t & 31u;                          // float4 slot within row
    const unsigned s_out = row & (NS - 1u);                  // seq index (S is pow2)
    const unsigned bh    = row >> 10;                        // batch*head index

    const int sidx = inv[s_out];  // wave-uniform -> scalar load

    size_t src_off;               // element offset of source row (same for K and V)
    const float *ksrc, *vsrc;
    if (sidx >= 0) {
        // Row was scatter-written: pull from the new values (B,H,S,D layout).
        src_off = ((size_t)bh * NS + (unsigned)sidx) * ND;
        ksrc = k_val + src_off;
        vsrc = v_val + src_off;
    } else {
        // Row untouched: pull from the original cache (B,H,BLOCK,D layout).
        src_off = ((size_t)bh * NBLOCK + s_out) * ND;
        ksrc = k_cache + src_off;
        vsrc = v_cache + src_off;
    }

    const size_t dst_off = (size_t)row * ND;                 // (B,H,S,D) flat

    f32x4 kx = __builtin_nontemporal_load((const f32x4*)ksrc + lane4); // b128 th:NT
    f32x4 vx = __builtin_nontemporal_load((const f32x4*)vsrc + lane4); // b128 th:NT

    __builtin_nontemporal_store(kx, (f32x4*)(out + dst_off)         + lane4); // K half
    __builtin_nontemporal_store(vx, (f32x4*)(out + ELEMS + dst_off) + lane4); // V half
}

// ---------------------------------------------------------------------------
// Launch: inputs in setup_inputs() order:
//   d_in[0] input_pos (int32, S)
//   d_in[1] k_val  (fp32, B*H*S*D)
//   d_in[2] v_val  (fp32, B*H*S*D)
//   d_in[3] k_cache(fp32, B*H*BLOCK*D)
//   d_in[4] v_cache(fp32, B*H*BLOCK*D)
// d_out = [k_out | v_out] flat, 2*B*H*S*D fp32.
// d_ws: 4 KB used for the inverse index.
// ---------------------------------------------------------------------------
extern "C" void kernel_launch(void* const* d_in, const int* in_sizes, int n_in,
                              void* d_out, int out_size, void* d_ws, size_t ws_size,
                              hipStream_t stream)
{
    const int*   input_pos = (const int*)  d_in[0];
    const float* k_val     = (const float*)d_in[1];
    const float* v_val     = (const float*)d_in[2];
    const float* k_cache   = (const float*)d_in[3];
    const float* v_cache   = (const float*)d_in[4];
    float*       out       = (float*)d_out;
    int*         inv       = (int*)d_ws;   // S ints = 4 KB scratch

    (void)in_sizes; (void)n_in; (void)out_size; (void)ws_size;

    // Build inverse scatter map (tiny; stream order gives init -> scatter -> gather).
    kv_init_inv <<<(NS + 255) / 256, 256, 0, stream>>>(inv);
    kv_build_inv<<<(NS + 255) / 256, 256, 0, stream>>>(input_pos, inv);

    // One-touch gather: 131072 waves, 8 waves (256 thr) per block, K+V per wave.
    kv_gather_rows<<<(unsigned)(ROWS * Q_PER_ROW / 256), 256, 0, stream>>>(
        k_val, v_val, k_cache, v_cache, inv, out);
}